// InteractionLayer_65438121722101
// MI455X (gfx1250) — compile-verified
//
#include <hip/hip_runtime.h>
#include <hip/hip_bf16.h>
#include <math.h>

// Problem constants (from reference)
#define FF 32
#define VV 100000
#define DD 16
#define BB 8192
#define PP 496          // 32*31/2
#define RR (BB * FF)    // 262144 rows of (b,f)
#define BN_EPS 0.001f

typedef float v2f __attribute__((ext_vector_type(2)));
typedef float v8f __attribute__((ext_vector_type(8)));

// ---------------------------------------------------------------------------
// Kernel 1: gather embeddings + per-field transforms.
//   For each row r=(b,f):  e = emb_table[f, X[r]]   (16 floats)
//     E[r]  = e                         (stored for pair kernel)
//     A[r]  = dot(w_fc, e)             (folds the 'add' branch)
//     U1[r] = W_cat[:, :16] @ e        (cat branch, via fp32 WMMA)
//     U2[r] = W_cat[:, 16:] @ e
//   One wave32 per 16-row tile; the 16x16 GEMMs run as 4 chained
//   V_WMMA_F32_16X16X4_F32 ops (K=16), full fp32 precision.
// ---------------------------------------------------------------------------
__global__ __launch_bounds__(256)
void k1_gather_transform(const int* __restrict__ X,
                         const float* __restrict__ emb,
                         const float* __restrict__ Wcat,
                         const float* __restrict__ wfc,
                         float* __restrict__ Ews,
                         float* __restrict__ Aws,
                         float* __restrict__ U1ws,
                         float* __restrict__ U2ws)
{
    const int lane = threadIdx.x & 31;
    const int wave = threadIdx.x >> 5;
    const int tile = blockIdx.x * 8 + wave;         // 16384 tiles total
    const int m  = lane & 15;                       // row within tile
    const int hi = lane >> 4;                       // half-wave selector
    const int r  = tile * 16 + m;                   // global (b,f) row
    const int f  = r & (FF - 1);

    const int idx = X[r];
    const float* row = emb + ((size_t)f * VV + (size_t)idx) * DD;

    // A fragments in ISA layout: lane(hi,m), vgpr pair kk -> K = 4*kk + 2*hi + {0,1}
    v2f afr[4];
    #pragma unroll
    for (int kk = 0; kk < 4; ++kk)
        afr[kk] = *(const v2f*)(row + kk * 4 + hi * 2);

    // Persist gathered row to workspace (each element written exactly once)
    #pragma unroll
    for (int kk = 0; kk < 4; ++kk)
        *(v2f*)(Ews + (size_t)r * DD + kk * 4 + hi * 2) = afr[kk];

    // a[r] = w_fc . e  (partial per half-wave, combine across hi)
    float part = 0.f;
    #pragma unroll
    for (int kk = 0; kk < 4; ++kk) {
        const int k0 = kk * 4 + hi * 2;
        part += afr[kk].x * wfc[k0] + afr[kk].y * wfc[k0 + 1];
    }
    part += __shfl_xor(part, 16, 32);
    if (hi == 0) Aws[r] = part;

    // B fragments: B[k][n] = Wcat[n][k] (and +16 col offset for U2)
    const int n = lane & 15;
    v2f b1[4], b2[4];
    #pragma unroll
    for (int kk = 0; kk < 4; ++kk) {
        const int k0 = kk * 4 + hi * 2;
        b1[kk] = *(const v2f*)(Wcat + n * 32 + k0);
        b2[kk] = *(const v2f*)(Wcat + n * 32 + 16 + k0);
    }

    v8f c1 = {}; v8f c2 = {};
    #pragma unroll
    for (int kk = 0; kk < 4; ++kk) {
        c1 = __builtin_amdgcn_wmma_f32_16x16x4_f32(false, afr[kk], false, b1[kk],
                                                   (short)0, c1, false, false);
        c2 = __builtin_amdgcn_wmma_f32_16x16x4_f32(false, afr[kk], false, b2[kk],
                                                   (short)0, c2, false, false);
    }

    // D layout: vgpr t, lane -> element (M = t + 8*hi, N = lane&15); coalesced stores
    #pragma unroll
    for (int t = 0; t < 8; ++t) {
        const int rr = tile * 16 + t + 8 * hi;
        U1ws[(size_t)rr * DD + n] = c1[t];
        U2ws[(size_t)rr * DD + n] = c2[t];
    }
}

// ---------------------------------------------------------------------------
// Kernel 2: per-(b,pair) scores. One wave per batch row, 8 rows per block.
// All per-b data staged in LDS (row stride 17 to spread banks).
// ---------------------------------------------------------------------------
__global__ __launch_bounds__(256)
void k2_pair_scores(const float* __restrict__ Ews,
                    const float* __restrict__ U1ws,
                    const float* __restrict__ U2ws,
                    const float* __restrict__ Aws,
                    const int* __restrict__ sel_type,
                    const float* __restrict__ Wgp,
                    const float* __restrict__ wfc,
                    float* __restrict__ S)
{
    __shared__ float sE [8][FF * 17];
    __shared__ float sU1[8][FF * 17];
    __shared__ float sU2[8][FF * 17];
    __shared__ float sA [8][FF];
    __shared__ int   sPair[PP];          // i | j<<5 | sel<<10
    __shared__ float sWfc[DD], sVgp[DD];

    const int tid  = threadIdx.x;
    const int lane = tid & 31;
    const int wave = tid >> 5;
    const int b    = blockIdx.x * 8 + wave;

    // Pair table (triu_indices order) + selected op
    for (int p = tid; p < PP; p += 256) {
        int i = 0, rem = p;
        while (rem >= (FF - 1) - i) { rem -= (FF - 1) - i; ++i; }
        const int j = i + 1 + rem;
        sPair[p] = i | (j << 5) | ((sel_type[p] & 3) << 10);
    }
    // w_fc and folded v_gp = W_gp^T w_fc
    if (tid < DD) {
        sWfc[tid] = wfc[tid];
        float acc = 0.f;
        #pragma unroll
        for (int e = 0; e < DD; ++e) acc += wfc[e] * Wgp[e * DD + tid];
        sVgp[tid] = acc;
    }

    // Stage this wave's batch row: E, U1, U2 (32x16 each), A (32)
    const size_t base = (size_t)b * FF * DD;
    for (int t = lane; t < FF * DD; t += 32) {
        const int rw = t >> 4, c = t & 15;
        sE [wave][rw * 17 + c] = Ews [base + t];
        sU1[wave][rw * 17 + c] = U1ws[base + t];
        sU2[wave][rw * 17 + c] = U2ws[base + t];
    }
    sA[wave][lane] = Aws[(size_t)b * FF + lane];
    __syncthreads();

    for (int p = lane; p < PP; p += 32) {
        const int pk  = sPair[p];
        const int i   = pk & 31;
        const int j   = (pk >> 5) & 31;
        const int sel = pk >> 10;
        float s;
        if (sel == 0) {
            s = sA[wave][i] + sA[wave][j];
        } else if (sel == 2) {
            const float* u1 = &sU1[wave][i * 17];
            const float* u2 = &sU2[wave][j * 17];
            float acc = 0.f;
            #pragma unroll
            for (int d = 0; d < DD; ++d)
                acc += sWfc[d] * fmaxf(u1[d] + u2[d], 0.f);
            s = acc;
        } else {
            const float* v  = (sel == 1) ? sWfc : sVgp;
            const float* e1 = &sE[wave][i * 17];
            const float* e2 = &sE[wave][j * 17];
            float acc = 0.f;
            #pragma unroll
            for (int d = 0; d < DD; ++d)
                acc += v[d] * e1[d] * e2[d];
            s = acc;
        }
        S[(size_t)b * PP + p] = s;
    }
}

// ---------------------------------------------------------------------------
// Kernel 3: per-pair mean/var over B, fold tanh(beta)*rsqrt(var+eps).
//   stats[p]     = c_p = tanh(beta_p) * rsqrt(var_p + eps)
//   stats[P + p] = c_p * mean_p
// ---------------------------------------------------------------------------
__global__ __launch_bounds__(256)
void k3_bn_stats(const float* __restrict__ S,
                 const float* __restrict__ beta,
                 float* __restrict__ stats)
{
    __shared__ float rs[256], rq[256];
    const int p = blockIdx.x, tid = threadIdx.x;
    float sum = 0.f, sq = 0.f;
    for (int b = tid; b < BB; b += 256) {
        const float v = S[(size_t)b * PP + p];
        sum += v; sq += v * v;
    }
    rs[tid] = sum; rq[tid] = sq;
    __syncthreads();
    for (int s = 128; s > 0; s >>= 1) {
        if (tid < s) { rs[tid] += rs[tid + s]; rq[tid] += rq[tid + s]; }
        __syncthreads();
    }
    if (tid == 0) {
        const float mean = rs[0] * (1.f / BB);
        float var = rq[0] * (1.f / BB) - mean * mean;
        var = fmaxf(var, 0.f);
        const float c = tanhf(beta[p]) * rsqrtf(var + BN_EPS);
        stats[p]      = c;
        stats[PP + p] = c * mean;
    }
}

// ---------------------------------------------------------------------------
// Kernel 4: out[b] = sum_p ( c_p * S[b,p] - c_p*mean_p )
// ---------------------------------------------------------------------------
__global__ __launch_bounds__(256)
void k4_output(const float* __restrict__ S,
               const float* __restrict__ stats,
               float* __restrict__ out)
{
    __shared__ float red[256];
    const int b = blockIdx.x, tid = threadIdx.x;
    float acc = 0.f;
    for (int p = tid; p < PP; p += 256)
        acc += stats[p] * S[(size_t)b * PP + p] - stats[PP + p];
    red[tid] = acc;
    __syncthreads();
    for (int s = 128; s > 0; s >>= 1) {
        if (tid < s) red[tid] += red[tid + s];
        __syncthreads();
    }
    if (tid == 0) out[b] = red[0];
}

// ---------------------------------------------------------------------------
extern "C" void kernel_launch(void* const* d_in, const int* in_sizes, int n_in,
                              void* d_out, int out_size, void* d_ws, size_t ws_size,
                              hipStream_t stream)
{
    const int*   X    = (const int*)  d_in[0];   // (B,F)
    const int*   sel  = (const int*)  d_in[1];   // (P,)
    const float* emb  = (const float*)d_in[2];   // (F,V,D)
    const float* Wcat = (const float*)d_in[3];   // (D,2D)
    const float* Wgp  = (const float*)d_in[4];   // (D,D)
    const float* wfc  = (const float*)d_in[5];   // (1,D)
    const float* beta = (const float*)d_in[6];   // (P,)
    float* out = (float*)d_out;

    // Workspace layout (floats)
    float* ws    = (float*)d_ws;
    float* Ews   = ws;                                   // R*16
    float* U1ws  = Ews  + (size_t)RR * DD;               // R*16
    float* U2ws  = U1ws + (size_t)RR * DD;               // R*16
    float* Aws   = U2ws + (size_t)RR * DD;               // R
    float* S     = Aws  + (size_t)RR;                    // B*P
    float* stats = S    + (size_t)BB * PP;               // 2*P

    k1_gather_transform<<<RR / 16 / 8, 256, 0, stream>>>(X, emb, Wcat, wfc,
                                                         Ews, Aws, U1ws, U2ws);
    k2_pair_scores<<<BB / 8, 256, 0, stream>>>(Ews, U1ws, U2ws, Aws,
                                               sel, Wgp, wfc, S);
    k3_bn_stats<<<PP, 256, 0, stream>>>(S, beta, stats);
    k4_output<<<BB, 256, 0, stream>>>(S, stats, out);
}